// Model_87119116632108
// MI455X (gfx1250) — compile-verified
//
#include <hip/hip_runtime.h>
#include <hip/hip_bf16.h>

// ---------------------------------------------------------------------------
// Problem constants (match reference)
// ---------------------------------------------------------------------------
#define N_CONST 131072
#define E_CONST 262144
#define L_CONST 16384
#define G_CONST 2048
#define D_CONST 300
#define NUM_LAYERS 5

typedef __attribute__((ext_vector_type(2))) float v2f;
typedef __attribute__((ext_vector_type(8))) float v8f;

// ---------------------------------------------------------------------------
// CDNA5 async global->LDS copy support (ASYNCcnt-tracked). Guarded so a
// toolchain without these builtins still compiles via the sync fallback.
// Builtin signatures (from compiler diagnostics):
//   b128: (int __vector(4) addrspace(1)*, int __vector(4) addrspace(3)*, Ii, Ii)
//   b32 : (int            addrspace(1)*, int            addrspace(3)*, Ii, Ii)
// ---------------------------------------------------------------------------
#if defined(__gfx1250__) &&                                                    \
    __has_builtin(__builtin_amdgcn_global_load_async_to_lds_b128) &&           \
    __has_builtin(__builtin_amdgcn_global_load_async_to_lds_b32) &&            \
    __has_builtin(__builtin_amdgcn_s_wait_asynccnt)
#define USE_ASYNC_LDS 1
#else
#define USE_ASYNC_LDS 0
#endif

#if USE_ASYNC_LDS
typedef int v4i_gcc __attribute__((vector_size(16)));

// Generic AMDGCN pointers to LDS carry the 0-based LDS offset in the low 32
// bits; generic pointers to global memory equal the global address.  Recover
// address-space- and pointee-typed pointers via integer round-trips.
template <typename T>
__device__ __forceinline__ __attribute__((address_space(1))) T*
as_global(const void* p) {
    return (__attribute__((address_space(1))) T*)(unsigned long long)(uintptr_t)p;
}
template <typename T>
__device__ __forceinline__ __attribute__((address_space(3))) T*
as_lds(const void* p) {
    return (__attribute__((address_space(3))) T*)(unsigned int)(uintptr_t)p;
}
#endif

// ---------------------------------------------------------------------------
// zero fill
// ---------------------------------------------------------------------------
__global__ void zero_kernel(float* __restrict__ p, long n) {
    long i = (long)blockIdx.x * blockDim.x + threadIdx.x;
    if (i < n) p[i] = 0.0f;
}

// ---------------------------------------------------------------------------
// node embedding: h[n,d] = x_emb1[x0[n],d] + x_emb2[x1[n],d]
// ---------------------------------------------------------------------------
__global__ void embed_kernel(const int* __restrict__ x,
                             const float* __restrict__ emb1,
                             const float* __restrict__ emb2,
                             float* __restrict__ h) {
    int idx = blockIdx.x * blockDim.x + threadIdx.x;   // over N*75
    if (idx >= N_CONST * 75) return;
    int n  = idx / 75;
    int c4 = (idx % 75) * 4;
    int x0 = x[2 * n];
    int x1 = x[2 * n + 1];
    float4 a = *(const float4*)(emb1 + (size_t)x0 * D_CONST + c4);
    float4 b = *(const float4*)(emb2 + (size_t)x1 * D_CONST + c4);
    float4 r = make_float4(a.x + b.x, a.y + b.y, a.z + b.z, a.w + b.w);
    *(float4*)(h + (size_t)n * D_CONST + c4) = r;
}

// ---------------------------------------------------------------------------
// message + scatter-add: agg[dst] += h[src] + edge_emb1[ea0] + edge_emb2[ea1]
// agg (157MB) fits in the 192MB L2 -> atomics stay on-chip.
// ---------------------------------------------------------------------------
__global__ void agg_kernel(const float* __restrict__ h,
                           const int* __restrict__ edge_index, // [2,E]
                           const int* __restrict__ edge_attr,  // [E,2]
                           const float* __restrict__ emb1,     // [6,D]
                           const float* __restrict__ emb2,     // [3,D]
                           float* __restrict__ agg) {
    int idx = blockIdx.x * blockDim.x + threadIdx.x;   // over (E+N)*75
    if (idx >= (E_CONST + N_CONST) * 75) return;
    int e  = idx / 75;
    int c4 = (idx % 75) * 4;
    int s, d, a0, a1;
    if (e < E_CONST) {
        s  = edge_index[e];
        d  = edge_index[E_CONST + e];
        a0 = edge_attr[2 * e];
        a1 = edge_attr[2 * e + 1];
    } else {
        s = d = e - E_CONST;
        a0 = 4; a1 = 0;
    }
    float4 hv = *(const float4*)(h    + (size_t)s  * D_CONST + c4);
    float4 e1 = *(const float4*)(emb1 + (size_t)a0 * D_CONST + c4);
    float4 e2 = *(const float4*)(emb2 + (size_t)a1 * D_CONST + c4);
    float* out = agg + (size_t)d * D_CONST + c4;
    atomicAdd(out + 0, hv.x + e1.x + e2.x);
    atomicAdd(out + 1, hv.y + e1.y + e2.y);
    atomicAdd(out + 2, hv.z + e1.z + e2.z);
    atomicAdd(out + 3, hv.w + e1.w + e2.w);
}

// ---------------------------------------------------------------------------
// fp32 WMMA GEMM:  C[M,N] = act(A[M,K] @ B[K,N] + bias[N])
// 256 threads = 8 waves; block tile 128x128; wave tile 32x64 (2x4 WMMA tiles);
// BK=32, double-buffered LDS, next tile staged with GLOBAL_LOAD_ASYNC_TO_LDS
// (ASYNCcnt) while the current tile feeds V_WMMA_F32_16X16X4_F32.
// Fragment layout per ISA: lanes 0-15 hold M/N rows with K={0,1} in the two
// VGPRs; lanes 16-31 hold K={2,3}.  A stored [m][k], B transposed to [n][k],
// stride 34 floats (8B-aligned v2f LDS loads).
// Tail tiles: partial-K tile staged synchronously with zero padding; lanes
// with n>=N are EXEC-predicated off the async load (their stale LDS columns
// only feed C columns that the epilogue discards).
// ---------------------------------------------------------------------------
#define BM 128
#define BNN 128
#define BK 32
#define LDS_S 34

__global__ __launch_bounds__(256) void gemm_bias_act(
        const float* __restrict__ A, const float* __restrict__ B,
        const float* __restrict__ bias, float* __restrict__ C,
        int M, int N, int K, int do_relu) {
    __shared__ float As[2][BM][LDS_S];
    __shared__ float Bs[2][BNN][LDS_S];

    const int tid  = threadIdx.x;
    const int bm   = blockIdx.x * BM;
    const int bn   = blockIdx.y * BNN;
    const int wid  = tid >> 5;
    const int lane = tid & 31;
    const int hl   = lane >> 4;       // wave half: K pair {0,1} vs {2,3}
    const int lrow = lane & 15;       // M (for A) / N (for B) within tile
    const int wm   = (wid & 3) * 32;  // 4 waves along M
    const int wn   = (wid >> 2) * 64; // 2 waves along N

    v8f acc[2][4];
    for (int i = 0; i < 2; ++i)
        for (int j = 0; j < 4; ++j)
            acc[i][j] = (v8f)(0.0f);

    const int T = (K + BK - 1) / BK;

    auto stage = [&](int b, int t) {
        const int k0 = t * BK;
        if (k0 + BK <= K) {              // full-K tile (uniform branch)
#if USE_ASYNC_LDS
            for (int j = 0; j < 4; ++j) {           // A: 4 x b128 per thread
                int li  = j * 256 + tid;
                int row = li >> 3;
                int c4  = (li & 7) * 4;
                __builtin_amdgcn_global_load_async_to_lds_b128(
                    as_global<v4i_gcc>(A + (size_t)(bm + row) * K + k0 + c4),
                    as_lds<v4i_gcc>(&As[b][row][c4]), 0, 0);
            }
            for (int j = 0; j < 16; ++j) {          // B: transpose scatter, b32
                int li = j * 256 + tid;
                int kk = li >> 7;                   // 0..31
                int nn = li & 127;
                int gn = bn + nn;
                if (gn < N)
                    __builtin_amdgcn_global_load_async_to_lds_b32(
                        as_global<int>(B + (size_t)(k0 + kk) * N + gn),
                        as_lds<int>(&Bs[b][nn][kk]), 0, 0);
            }
#else
            for (int j = 0; j < 4; ++j) {
                int li  = j * 256 + tid;
                int row = li >> 3;
                int c4  = (li & 7) * 4;
                *(float4*)&As[b][row][c4] =
                    *(const float4*)(A + (size_t)(bm + row) * K + k0 + c4);
            }
            for (int j = 0; j < 16; ++j) {
                int li = j * 256 + tid;
                int kk = li >> 7;
                int nn = li & 127;
                int gn = bn + nn;
                Bs[b][nn][kk] = (gn < N) ? B[(size_t)(k0 + kk) * N + gn] : 0.0f;
            }
#endif
        } else {                          // partial-K tail: sync, zero-padded
            for (int j = 0; j < 4; ++j) {
                int li  = j * 256 + tid;
                int row = li >> 3;
                int c4  = (li & 7) * 4;
                int gk  = k0 + c4;
                float4 v = make_float4(0.f, 0.f, 0.f, 0.f);
                if (gk < K)   // K % 4 == 0 -> float4 fully in or out of range
                    v = *(const float4*)(A + (size_t)(bm + row) * K + gk);
                *(float4*)&As[b][row][c4] = v;
            }
            for (int j = 0; j < 16; ++j) {
                int li = j * 256 + tid;
                int kk = li >> 7;
                int nn = li & 127;
                int gk = k0 + kk, gn = bn + nn;
                float v = 0.f;
                if (gk < K && gn < N) v = B[(size_t)gk * N + gn];
                Bs[b][nn][kk] = v;
            }
        }
    };

    stage(0, 0);

    for (int t = 0; t < T; ++t) {
#if USE_ASYNC_LDS
        __builtin_amdgcn_s_wait_asynccnt(0);   // our stage(t) writes landed
#endif
        __syncthreads();                       // everyone's stage(t) visible;
                                               // everyone done reading buf t-1
        if (t + 1 < T) stage((t + 1) & 1, t + 1);

        const int b = t & 1;
        for (int k = 0; k < BK; k += 4) {
            int kk = k + hl * 2;
            v2f a0 = *(const v2f*)&As[b][wm + lrow][kk];
            v2f a1 = *(const v2f*)&As[b][wm + 16 + lrow][kk];
            v2f b0 = *(const v2f*)&Bs[b][wn + lrow][kk];
            v2f b1 = *(const v2f*)&Bs[b][wn + 16 + lrow][kk];
            v2f b2 = *(const v2f*)&Bs[b][wn + 32 + lrow][kk];
            v2f b3 = *(const v2f*)&Bs[b][wn + 48 + lrow][kk];
            acc[0][0] = __builtin_amdgcn_wmma_f32_16x16x4_f32(false, a0, false, b0,
                          (short)0, acc[0][0], false, false);
            acc[0][1] = __builtin_amdgcn_wmma_f32_16x16x4_f32(false, a0, false, b1,
                          (short)0, acc[0][1], false, false);
            acc[0][2] = __builtin_amdgcn_wmma_f32_16x16x4_f32(false, a0, false, b2,
                          (short)0, acc[0][2], false, false);
            acc[0][3] = __builtin_amdgcn_wmma_f32_16x16x4_f32(false, a0, false, b3,
                          (short)0, acc[0][3], false, false);
            acc[1][0] = __builtin_amdgcn_wmma_f32_16x16x4_f32(false, a1, false, b0,
                          (short)0, acc[1][0], false, false);
            acc[1][1] = __builtin_amdgcn_wmma_f32_16x16x4_f32(false, a1, false, b1,
                          (short)0, acc[1][1], false, false);
            acc[1][2] = __builtin_amdgcn_wmma_f32_16x16x4_f32(false, a1, false, b2,
                          (short)0, acc[1][2], false, false);
            acc[1][3] = __builtin_amdgcn_wmma_f32_16x16x4_f32(false, a1, false, b3,
                          (short)0, acc[1][3], false, false);
        }
    }

    // epilogue: C/D layout -> element (vgpr v, lane): m = hl*8+v, n = lrow
    for (int mt = 0; mt < 2; ++mt)
        for (int nt = 0; nt < 4; ++nt) {
            int gn = bn + wn + nt * 16 + lrow;
            if (gn >= N) continue;
            float bv = bias[gn];
            for (int v = 0; v < 8; ++v) {
                int gm = bm + wm + mt * 16 + hl * 8 + v;
                if (gm >= M) continue;
                float val = acc[mt][nt][v] + bv;
                if (do_relu) val = fmaxf(val, 0.0f);
                C[(size_t)gm * N + gn] = val;
            }
        }
}

// ---------------------------------------------------------------------------
// BatchNorm: column sums / sumsq (thread d per column, block strides rows)
// ---------------------------------------------------------------------------
__global__ void bn_stats(const float* __restrict__ h2, float* __restrict__ stats) {
    int d = threadIdx.x;
    if (d >= D_CONST) return;
    float s = 0.f, ss = 0.f;
    for (int r = blockIdx.x; r < N_CONST; r += gridDim.x) {
        float v = h2[(size_t)r * D_CONST + d];
        s += v; ss += v * v;
    }
    atomicAdd(&stats[d], s);
    atomicAdd(&stats[D_CONST + d], ss);
}

__global__ void bn_finalize(const float* __restrict__ stats,
                            const float* __restrict__ gamma,
                            const float* __restrict__ beta,
                            float* __restrict__ scsh) {
    int d = threadIdx.x;
    if (d >= D_CONST) return;
    float invN = 1.0f / (float)N_CONST;
    float mean = stats[d] * invN;
    float var  = stats[D_CONST + d] * invN - mean * mean;
    float sc   = gamma[d] * rsqrtf(var + 1e-5f);
    scsh[d]           = sc;
    scsh[D_CONST + d] = beta[d] - mean * sc;
}

__global__ void bn_apply(const float* __restrict__ h2, const float* __restrict__ scsh,
                         float* __restrict__ h, int do_relu) {
    long idx = (long)blockIdx.x * blockDim.x + threadIdx.x; // over N*D
    if (idx >= (long)N_CONST * D_CONST) return;
    int d = (int)(idx % D_CONST);
    float v = h2[idx] * scsh[d] + scsh[D_CONST + d];
    if (do_relu) v = fmaxf(v, 0.0f);
    h[idx] = v;
}

// ---------------------------------------------------------------------------
// segment means over SORTED segment ids: one block per segment, redundant
// binary search for bounds (scalar broadcast loads), no atomics.
// ---------------------------------------------------------------------------
__global__ void lower_pool(const float* __restrict__ h, const int* __restrict__ lb,
                           float* __restrict__ lower) {
    int l = blockIdx.x;
    int lo = 0, hi = N_CONST;
    while (lo < hi) { int m = (lo + hi) >> 1; if (lb[m] < l) lo = m + 1; else hi = m; }
    int start = lo;
    hi = N_CONST;
    while (lo < hi) { int m = (lo + hi) >> 1; if (lb[m] < l + 1) lo = m + 1; else hi = m; }
    int end = lo;
    int d = threadIdx.x;
    if (d >= D_CONST) return;
    float s = 0.f;
    for (int i = start; i < end; ++i) s += h[(size_t)i * D_CONST + d];
    int cnt = end - start;
    lower[(size_t)l * D_CONST + d] = s / (float)(cnt > 0 ? cnt : 1);
}

__global__ void upper_pool(const float* __restrict__ lower, const int* __restrict__ ub,
                           float* __restrict__ p0, float* __restrict__ p1) {
    int g = blockIdx.x;
    int lo = 0, hi = L_CONST;
    while (lo < hi) { int m = (lo + hi) >> 1; if (ub[m] < g) lo = m + 1; else hi = m; }
    int start = lo;
    hi = L_CONST;
    while (lo < hi) { int m = (lo + hi) >> 1; if (ub[m] < g + 1) lo = m + 1; else hi = m; }
    int end = lo;
    int d = threadIdx.x;
    if (d >= D_CONST) return;
    float s0 = 0.f, s1 = 0.f;
    for (int i = start; i < end; ++i) {
        s0 += lower[(size_t)i * D_CONST + d];
        int j = (i == 0) ? (L_CONST - 1) : (i - 1);   // roll(lower, 1)
        s1 += lower[(size_t)j * D_CONST + d];
    }
    int cnt = end - start;
    float inv = 1.0f / (float)(cnt > 0 ? cnt : 1);
    p0[(size_t)g * D_CONST + d] = s0 * inv;
    p1[(size_t)g * D_CONST + d] = s1 * inv;
}

// ---------------------------------------------------------------------------
// tiny classifier head: hh = relu(pooled @ W1 + b1) ; logits = hh @ w2 + b2
// ---------------------------------------------------------------------------
__global__ void cls_hidden(const float* __restrict__ pooled, const float* __restrict__ w1,
                           const float* __restrict__ b1, float* __restrict__ hh, int rows) {
    int idx = blockIdx.x * blockDim.x + threadIdx.x;  // over rows*D
    if (idx >= rows * D_CONST) return;
    int g = idx / D_CONST, d = idx % D_CONST;
    const float* pr = pooled + (size_t)g * D_CONST;
    float acc = b1[d];
    for (int k = 0; k < D_CONST; ++k) acc += pr[k] * w1[(size_t)k * D_CONST + d];
    hh[idx] = fmaxf(acc, 0.0f);
}

__global__ void cls_logits(const float* __restrict__ hh, const float* __restrict__ w2,
                           const float* __restrict__ b2, float* __restrict__ out, int rows) {
    int g = blockIdx.x * blockDim.x + threadIdx.x;
    if (g >= rows) return;
    const float* r = hh + (size_t)g * D_CONST;
    float acc = b2[0];
    for (int d = 0; d < D_CONST; ++d) acc += r[d] * w2[d];
    out[g] = acc;
}

// ---------------------------------------------------------------------------
// host-side orchestration
// ---------------------------------------------------------------------------
extern "C" void kernel_launch(void* const* d_in, const int* in_sizes, int n_in,
                              void* d_out, int out_size, void* d_ws, size_t ws_size,
                              hipStream_t stream) {
    const int*   x          = (const int*)d_in[0];
    const int*   edge_index = (const int*)d_in[1];
    const int*   edge_attr  = (const int*)d_in[2];
    const int*   lower_b    = (const int*)d_in[3];
    const int*   upper_b    = (const int*)d_in[4];
    const float* x_emb1     = (const float*)d_in[5];
    const float* x_emb2     = (const float*)d_in[6];
    const float* edge_emb1  = (const float*)d_in[7];   // [5,6,D]
    const float* edge_emb2  = (const float*)d_in[8];   // [5,3,D]
    const float* mlp_w1     = (const float*)d_in[9];   // [5,D,2D]
    const float* mlp_b1     = (const float*)d_in[10];  // [5,2D]
    const float* mlp_w2     = (const float*)d_in[11];  // [5,2D,D]
    const float* mlp_b2     = (const float*)d_in[12];  // [5,D]
    const float* bn_gamma   = (const float*)d_in[13];
    const float* bn_beta    = (const float*)d_in[14];
    const float* cls_w1     = (const float*)d_in[15];
    const float* cls_b1     = (const float*)d_in[16];
    const float* cls_w2     = (const float*)d_in[17];
    const float* cls_b2     = (const float*)d_in[18];
    float* out = (float*)d_out;

    // workspace layout (floats)
    float* ws = (float*)d_ws;
    const size_t ND  = (size_t)N_CONST * D_CONST;       // 39.3M
    const size_t N2D = (size_t)N_CONST * 2 * D_CONST;   // 78.6M
    float* h      = ws;                                 // [N, D]
    float* aggh2  = h + ND;                             // agg, then reused as h2
    float* z      = aggh2 + ND;                         // [N, 2D]
    float* lower  = z + N2D;                            // [L, D]
    float* pooled = lower + (size_t)L_CONST * D_CONST;  // [2G, D] (p0 ++ p1)
    float* hh     = pooled + (size_t)2 * G_CONST * D_CONST; // [2G, D]
    float* stats  = hh + (size_t)2 * G_CONST * D_CONST; // [2D]
    float* scsh   = stats + 2 * D_CONST;                // [2D]

    const int TB = 256;

    // node embedding
    {
        int tot = N_CONST * 75;
        embed_kernel<<<(tot + TB - 1) / TB, TB, 0, stream>>>(x, x_emb1, x_emb2, h);
    }

    for (int l = 0; l < NUM_LAYERS; ++l) {
        // agg = 0
        zero_kernel<<<(int)((ND + TB - 1) / TB), TB, 0, stream>>>(aggh2, (long)ND);
        // scatter messages
        {
            int tot = (E_CONST + N_CONST) * 75;
            agg_kernel<<<(tot + TB - 1) / TB, TB, 0, stream>>>(
                h, edge_index, edge_attr,
                edge_emb1 + (size_t)l * 6 * D_CONST,
                edge_emb2 + (size_t)l * 3 * D_CONST, aggh2);
        }
        // z = relu(agg @ W1 + b1)   [N,300] x [300,600]
        {
            dim3 grid(N_CONST / BM, (2 * D_CONST + BNN - 1) / BNN);
            gemm_bias_act<<<grid, TB, 0, stream>>>(
                aggh2, mlp_w1 + (size_t)l * D_CONST * 2 * D_CONST,
                mlp_b1 + (size_t)l * 2 * D_CONST, z,
                N_CONST, 2 * D_CONST, D_CONST, 1);
        }
        // h2 = z @ W2 + b2          [N,600] x [600,300]  (h2 overwrites agg)
        {
            dim3 grid(N_CONST / BM, (D_CONST + BNN - 1) / BNN);
            gemm_bias_act<<<grid, TB, 0, stream>>>(
                z, mlp_w2 + (size_t)l * 2 * D_CONST * D_CONST,
                mlp_b2 + (size_t)l * D_CONST, aggh2,
                N_CONST, D_CONST, 2 * D_CONST, 0);
        }
        // BatchNorm (training mode, biased var) + relu (except last layer)
        zero_kernel<<<1, 2 * D_CONST, 0, stream>>>(stats, 2 * D_CONST);
        bn_stats<<<1024, 320, 0, stream>>>(aggh2, stats);
        bn_finalize<<<1, 320, 0, stream>>>(stats, bn_gamma + (size_t)l * D_CONST,
                                           bn_beta + (size_t)l * D_CONST, scsh);
        {
            long tot = (long)ND;
            bn_apply<<<(int)((tot + TB - 1) / TB), TB, 0, stream>>>(
                aggh2, scsh, h, (l < NUM_LAYERS - 1) ? 1 : 0);
        }
    }

    // hierarchical pooling (sorted segments -> block-per-segment, no atomics)
    lower_pool<<<L_CONST, 320, 0, stream>>>(h, lower_b, lower);
    upper_pool<<<G_CONST, 320, 0, stream>>>(lower, upper_b,
                                            pooled, pooled + (size_t)G_CONST * D_CONST);

    // classifier on both heads at once (rows = 2G)
    {
        int rows = 2 * G_CONST;
        int tot = rows * D_CONST;
        cls_hidden<<<(tot + TB - 1) / TB, TB, 0, stream>>>(pooled, cls_w1, cls_b1, hh, rows);
        cls_logits<<<(rows + TB - 1) / TB, TB, 0, stream>>>(hh, cls_w2, cls_b2, out, rows);
    }
}